// Aligned3DUpsampleConcat_3496103379586
// MI455X (gfx1250) — compile-verified
//
#include <hip/hip_runtime.h>

#define NVOX (64 * 64 * 64)   // 262144 voxels
#define FEAT 64
#define KDIM 128
#define TILES 8               // voxel-tiles (of 16) per GEMM workgroup

typedef __attribute__((ext_vector_type(2))) float v2f;
typedef __attribute__((ext_vector_type(8))) float v8f;

// ---------------------------------------------------------------------------
// Kernel 1: trilinear 2x upsample of lr [64,32,32,32] -> hs [64,64,64,64]
// PyTorch align_corners=False: src = 0.5*dst - 0.25, clamped at 0.
// ---------------------------------------------------------------------------
__device__ __forceinline__ void up_axis(int i, int& i0, int& i1, float& f) {
    float c = 0.5f * (float)i - 0.25f;
    c = fmaxf(c, 0.0f);
    i0 = (int)c;                 // floor (c >= 0)
    f = c - (float)i0;
    i1 = i0 + 1;
    if (i1 > 31) i1 = 31;
}

__global__ __launch_bounds__(256) void upsample_k(const float* __restrict__ lr,
                                                  float* __restrict__ hs) {
    int tid = blockIdx.x * 256 + threadIdx.x;      // c * NVOX + v
    int v = tid & (NVOX - 1);
    int c = tid >> 18;
    int wi = v & 63, hi = (v >> 6) & 63, di = v >> 12;

    int x0, x1, y0, y1, z0, z1;
    float fx, fy, fz;
    up_axis(wi, x0, x1, fx);
    up_axis(hi, y0, y1, fy);
    up_axis(di, z0, z1, fz);

    const float* b = lr + (size_t)c * 32768;
    float v000 = b[(z0 * 32 + y0) * 32 + x0], v001 = b[(z0 * 32 + y0) * 32 + x1];
    float v010 = b[(z0 * 32 + y1) * 32 + x0], v011 = b[(z0 * 32 + y1) * 32 + x1];
    float v100 = b[(z1 * 32 + y0) * 32 + x0], v101 = b[(z1 * 32 + y0) * 32 + x1];
    float v110 = b[(z1 * 32 + y1) * 32 + x0], v111 = b[(z1 * 32 + y1) * 32 + x1];

    float c00 = v000 * (1.f - fx) + v001 * fx;
    float c01 = v010 * (1.f - fx) + v011 * fx;
    float c10 = v100 * (1.f - fx) + v101 * fx;
    float c11 = v110 * (1.f - fx) + v111 * fx;
    float c0 = c00 * (1.f - fy) + c01 * fy;
    float c1 = c10 * (1.f - fy) + c11 * fy;
    hs[(size_t)c * NVOX + v] = c0 * (1.f - fz) + c1 * fz;
}

// ---------------------------------------------------------------------------
// Kernel 2: x = w1 @ concat([hr, hs])  (1x1 conv, K=128, 64 out channels)
// 128 threads = 4 waves; wave w owns out-channels [16w,16w+16).
// A (w1) fragment is REGISTER-resident (loaded once per block as float2s,
// ISA 16x4 layout: lane L holds M=L%16, K = k0 + vg + 2*(L/16)).
// Block iterates over TILES voxel-tiles of 16, staging only the 8 KB B tile
// in LDS per tile. 32x V_WMMA_F32_16X16X4_F32 per tile per wave.
// D layout: VGPR r -> M = r + 8*(L/16), N = L%16.
// ---------------------------------------------------------------------------
__global__ __launch_bounds__(128) void gemm1_k(const float* __restrict__ hr,
                                               const float* __restrict__ hs,
                                               const float* __restrict__ w1,
                                               float* __restrict__ xo) {
    __shared__ float sB[KDIM * 16];     // 8 KB
    int t = threadIdx.x;
    int wave = t >> 5, lane = t & 31;
    int half = lane >> 4, ln = lane & 15;
    int o0 = wave * 16;
    int row = o0 + ln;

    v2f a_frag[32];
#pragma unroll
    for (int s = 0; s < 32; ++s) {
        const float* p = w1 + row * KDIM + 4 * s + 2 * half;  // 8B aligned
        a_frag[s][0] = p[0];
        a_frag[s][1] = p[1];
    }

    for (int tile = 0; tile < TILES; ++tile) {
        int v0 = (blockIdx.x * TILES + tile) * 16;
        __syncthreads();   // sB safe to overwrite
        for (int e = t; e < KDIM * 16; e += 128) {
            int k = e >> 4, j = e & 15;
            sB[e] = (k < FEAT) ? hr[(size_t)k * NVOX + v0 + j]
                               : hs[(size_t)(k - FEAT) * NVOX + v0 + j];
        }
        __syncthreads();

        v8f acc = {};
#pragma unroll
        for (int k0 = 0; k0 < KDIM; k0 += 4) {
            v2f b;
            b[0] = sB[(k0 + 0 + 2 * half) * 16 + ln];
            b[1] = sB[(k0 + 1 + 2 * half) * 16 + ln];
            acc = __builtin_amdgcn_wmma_f32_16x16x4_f32(
                false, a_frag[k0 / 4], false, b, (short)0, acc, false, false);
        }
#pragma unroll
        for (int r = 0; r < 8; ++r) {
            int o = o0 + r + 8 * half;
            xo[(size_t)o * NVOX + v0 + ln] = acc[r];
        }
    }
}

// ---------------------------------------------------------------------------
// Kernel 3: group-norm statistics. 16 groups x 4 contiguous channels.
// Deterministic fixed-order tree reduction (no float atomics).
// ---------------------------------------------------------------------------
__global__ __launch_bounds__(256) void gn_reduce_k(const float* __restrict__ x,
                                                   float* __restrict__ stats) {
    int g = blockIdx.x;
    const float* p = x + (size_t)g * 4 * NVOX;
    float s = 0.f, s2 = 0.f;
    for (int i = threadIdx.x; i < 4 * NVOX; i += 256) {
        float v = p[i];
        s += v;
        s2 += v * v;
    }
    __shared__ float rs[256], rs2[256];
    rs[threadIdx.x] = s;
    rs2[threadIdx.x] = s2;
    __syncthreads();
    for (int off = 128; off > 0; off >>= 1) {
        if (threadIdx.x < off) {
            rs[threadIdx.x] += rs[threadIdx.x + off];
            rs2[threadIdx.x] += rs2[threadIdx.x + off];
        }
        __syncthreads();
    }
    if (threadIdx.x == 0) {
        float n = 4.0f * (float)NVOX;
        float mu = rs[0] / n;
        float var = rs2[0] / n - mu * mu;
        stats[2 * g] = mu;
        stats[2 * g + 1] = rsqrtf(var + 1e-5f);
    }
}

// ---------------------------------------------------------------------------
// Kernel 4: delta = conv3x3x3(group_norm(x)) + b2   (64 -> 3 channels, SAME)
// GN folded as per-channel affine a*x+b (zero padding applies AFTER norm).
// Tile 8x8x4 voxels per 256-thread block, halo tile (10x10x6) + w2 in LDS.
// ---------------------------------------------------------------------------
__global__ __launch_bounds__(256) void conv_delta_k(const float* __restrict__ x,
                                                    const float* __restrict__ stats,
                                                    const float* __restrict__ gamma,
                                                    const float* __restrict__ beta,
                                                    const float* __restrict__ w2,
                                                    const float* __restrict__ b2,
                                                    float* __restrict__ delta) {
    __shared__ float sw[3 * FEAT * 27];    // 20.7 KB
    __shared__ float tile[6 * 10 * 10];    // 2.4 KB
    int t = threadIdx.x;
    for (int i = t; i < 3 * FEAT * 27; i += 256) sw[i] = w2[i];

    int tx = t & 7, ty = (t >> 3) & 7, tz = t >> 6;   // 8 x 8 x 4
    int wx0 = (blockIdx.x & 7) * 8;
    int hy0 = ((blockIdx.x >> 3) & 7) * 8;
    int dz0 = (blockIdx.x >> 6) * 4;

    float acc0 = 0.f, acc1 = 0.f, acc2 = 0.f;
    for (int c = 0; c < FEAT; ++c) {
        __syncthreads();   // also covers initial sw load
        float a = stats[2 * (c >> 2) + 1] * gamma[c];
        float bb = beta[c] - stats[2 * (c >> 2)] * a;
        for (int i = t; i < 600; i += 256) {
            int lx = i % 10, ly = (i / 10) % 10, lz = i / 100;
            int gx = wx0 - 1 + lx, gy = hy0 - 1 + ly, gz = dz0 - 1 + lz;
            float val = 0.f;
            if (gx >= 0 && gx < 64 && gy >= 0 && gy < 64 && gz >= 0 && gz < 64)
                val = a * x[(size_t)c * NVOX + (gz << 12) + (gy << 6) + gx] + bb;
            tile[i] = val;
        }
        __syncthreads();
#pragma unroll
        for (int kd = 0; kd < 3; ++kd)
#pragma unroll
            for (int kh = 0; kh < 3; ++kh)
#pragma unroll
                for (int kw = 0; kw < 3; ++kw) {
                    float v = tile[(tz + kd) * 100 + (ty + kh) * 10 + (tx + kw)];
                    int wi_ = kd * 9 + kh * 3 + kw;
                    acc0 += sw[(0 * FEAT + c) * 27 + wi_] * v;
                    acc1 += sw[(1 * FEAT + c) * 27 + wi_] * v;
                    acc2 += sw[(2 * FEAT + c) * 27 + wi_] * v;
                }
    }
    int v = ((dz0 + tz) << 12) + ((hy0 + ty) << 6) + (wx0 + tx);
    delta[0 * NVOX + v] = acc0 + b2[0];
    delta[1 * NVOX + v] = acc1 + b2[1];
    delta[2 * NVOX + v] = acc2 + b2[2];
}

// ---------------------------------------------------------------------------
// Kernel 5: grid_sample(hs) -> aligned; merged = w3 @ [aligned, hs] + b3,
// written into the second half of the output. Same register-resident-A
// multi-tile structure as gemm1; phase 1 builds the K x 16 B-tile in LDS
// via an 8-corner zero-padded trilinear gather.
// ---------------------------------------------------------------------------
__global__ __launch_bounds__(128) void final_k(const float* __restrict__ hs,
                                               const float* __restrict__ delta,
                                               const float* __restrict__ w3,
                                               const float* __restrict__ b3,
                                               float* __restrict__ out) {
    __shared__ float sB[KDIM * 16];     // 8 KB
    int t = threadIdx.x;
    int wave = t >> 5, lane = t & 31;
    int half = lane >> 4, ln = lane & 15;
    int o0 = wave * 16;
    int row = o0 + ln;

    v2f a_frag[32];
#pragma unroll
    for (int s = 0; s < 32; ++s) {
        const float* p = w3 + row * KDIM + 4 * s + 2 * half;
        a_frag[s][0] = p[0];
        a_frag[s][1] = p[1];
    }
    float biasv[8];
#pragma unroll
    for (int r = 0; r < 8; ++r) biasv[r] = b3[o0 + r + 8 * half];

    int j = t & 15;       // owned B-tile column
    int krow = t >> 4;    // 0..7 -> rows krow + 8*i

    for (int tile = 0; tile < TILES; ++tile) {
        int v0 = (blockIdx.x * TILES + tile) * 16;
        int v = v0 + j;
        int wi = v & 63, hi = (v >> 6) & 63, di = v >> 12;

        // --- grid + sample geometry for column j ---
        const float step = 2.0f / 63.0f;
        float gx = -1.f + (float)wi * step + delta[v] * (1.f / 64.f);
        float gy = -1.f + (float)hi * step + delta[NVOX + v] * (1.f / 64.f);
        float gz = -1.f + (float)di * step + delta[2 * NVOX + v] * (1.f / 64.f);
        float ix = ((gx + 1.f) * 64.f - 1.f) * 0.5f;
        float iy = ((gy + 1.f) * 64.f - 1.f) * 0.5f;
        float iz = ((gz + 1.f) * 64.f - 1.f) * 0.5f;
        float x0f = floorf(ix), y0f = floorf(iy), z0f = floorf(iz);
        float fx = ix - x0f, fy = iy - y0f, fz = iz - z0f;
        int x0 = (int)x0f, y0 = (int)y0f, z0 = (int)z0f;

        int idx8[8];
        float wgt8[8];
        int q = 0;
#pragma unroll
        for (int dz_ = 0; dz_ < 2; ++dz_)
#pragma unroll
            for (int dy_ = 0; dy_ < 2; ++dy_)
#pragma unroll
                for (int dx_ = 0; dx_ < 2; ++dx_) {
                    int xc = x0 + dx_, yc = y0 + dy_, zc = z0 + dz_;
                    float w = (dx_ ? fx : 1.f - fx) * (dy_ ? fy : 1.f - fy) *
                              (dz_ ? fz : 1.f - fz);
                    bool valid = (xc >= 0) && (xc < 64) && (yc >= 0) &&
                                 (yc < 64) && (zc >= 0) && (zc < 64);
                    int xl = xc < 0 ? 0 : (xc > 63 ? 63 : xc);
                    int yl = yc < 0 ? 0 : (yc > 63 ? 63 : yc);
                    int zl = zc < 0 ? 0 : (zc > 63 ? 63 : zc);
                    idx8[q] = (zl << 12) + (yl << 6) + xl;
                    wgt8[q] = valid ? w : 0.f;
                    ++q;
                }

        __syncthreads();   // sB safe to overwrite
        for (int i = 0; i < 16; ++i) {
            int k = krow + 8 * i;
            float val;
            if (k < FEAT) {
                const float* hc = hs + (size_t)k * NVOX;
                val = 0.f;
#pragma unroll
                for (int qq = 0; qq < 8; ++qq) val += wgt8[qq] * hc[idx8[qq]];
            } else {
                val = hs[(size_t)(k - FEAT) * NVOX + v];
            }
            sB[k * 16 + j] = val;
        }
        __syncthreads();

        // --- WMMA GEMM with w3 ---
        v8f acc = {};
#pragma unroll
        for (int k0 = 0; k0 < KDIM; k0 += 4) {
            v2f b;
            b[0] = sB[(k0 + 0 + 2 * half) * 16 + ln];
            b[1] = sB[(k0 + 1 + 2 * half) * 16 + ln];
            acc = __builtin_amdgcn_wmma_f32_16x16x4_f32(
                false, a_frag[k0 / 4], false, b, (short)0, acc, false, false);
        }
#pragma unroll
        for (int r = 0; r < 8; ++r) {
            int o = o0 + r + 8 * half;
            out[(size_t)FEAT * NVOX + (size_t)o * NVOX + v0 + ln] =
                acc[r] + biasv[r];
        }
    }
}

// ---------------------------------------------------------------------------
// Launch
// ---------------------------------------------------------------------------
extern "C" void kernel_launch(void* const* d_in, const int* in_sizes, int n_in,
                              void* d_out, int out_size, void* d_ws, size_t ws_size,
                              hipStream_t stream) {
    const float* hr    = (const float*)d_in[0];
    const float* lr    = (const float*)d_in[1];
    const float* w1    = (const float*)d_in[2];
    const float* gamma = (const float*)d_in[3];
    const float* beta  = (const float*)d_in[4];
    const float* w2    = (const float*)d_in[5];
    const float* b2    = (const float*)d_in[6];
    const float* w3    = (const float*)d_in[7];
    const float* b3    = (const float*)d_in[8];
    float* out = (float*)d_out;

    float* ws    = (float*)d_ws;
    float* hs    = ws;                               // 64 MB
    float* x     = hs + (size_t)FEAT * NVOX;         // 64 MB
    float* delta = x + (size_t)FEAT * NVOX;          // 3 MB
    float* stats = delta + (size_t)3 * NVOX;         // 128 B

    // output first half = hr (channel concat)
    hipMemcpyAsync(out, hr, (size_t)FEAT * NVOX * sizeof(float),
                   hipMemcpyDeviceToDevice, stream);

    upsample_k<<<(FEAT * NVOX) / 256, 256, 0, stream>>>(lr, hs);
    gemm1_k<<<NVOX / (16 * TILES), 128, 0, stream>>>(hr, hs, w1, x);
    gn_reduce_k<<<16, 256, 0, stream>>>(x, stats);
    conv_delta_k<<<1024, 256, 0, stream>>>(x, stats, gamma, beta, w2, b2, delta);
    final_k<<<NVOX / (16 * TILES), 128, 0, stream>>>(hs, delta, w3, b3, out);
}